// BlazeFace_54606214201958
// MI455X (gfx1250) — compile-verified
//
#include <hip/hip_runtime.h>

// fp32 WMMA operand types (wave32): A/B = 2 VGPRs, C/D = 8 VGPRs.
typedef __attribute__((ext_vector_type(2))) float v2f;
typedef __attribute__((ext_vector_type(8))) float v8f;
// Tensor-DMA descriptor groups (clang-23 6-arg builtin form).
typedef __attribute__((ext_vector_type(4))) unsigned int v4u;
typedef __attribute__((ext_vector_type(8))) int v8i;
typedef __attribute__((ext_vector_type(4))) int v4i;

#if __has_builtin(__builtin_amdgcn_tensor_load_to_lds) && \
    __has_builtin(__builtin_amdgcn_s_wait_tensorcnt)
#define HAVE_TDM 1
#else
#define HAVE_TDM 0
#endif

static inline int cdiv_h(int a, int b) { return (a + b - 1) / b; }

// ---------------------------------------------------------------------------
// TDM: 1-D contiguous copy of `total` fp32 elements global -> LDS.
// D# layout per CDNA5 ISA §8.3/§8.4: group0 = {flags, lds_addr, global_addr,
// type=2}; group1 = {data_size=4B, tensor_dim0=total, tensor_dim1=1,
// tile_dim0=total, tile_dim1=1, stride0=total}. Groups 2/3 (+extra group of
// the 6-arg builtin) zero: tensor is <=2-D. Issued by one wave; EXEC ignored;
// completion via TENSORcnt.
// ---------------------------------------------------------------------------
#if HAVE_TDM
__device__ __forceinline__ void tdm_load_1d(const float* gsrc, float* lds_dst,
                                            unsigned total) {
  unsigned lds = (unsigned)(unsigned long long)(void*)lds_dst; // LDS byte offset
  unsigned long long ga = (unsigned long long)(void*)gsrc;
  v4u g0;
  g0[0] = 1u;                                          // count=1, user descriptor
  g0[1] = lds;                                         // lds_addr (bytes)
  g0[2] = (unsigned)(ga & 0xFFFFFFFFu);                // global_addr[31:0]
  g0[3] = (unsigned)((ga >> 32) & 0x01FFFFFFu) | 0x80000000u; // addr[56:32]|type=2
  v8i g1;
  g1[0] = (int)(2u << 16);                             // data_size = 4 bytes
  g1[1] = (int)((total & 0xFFFFu) << 16);              // tensor_dim0[15:0]
  g1[2] = (int)((total >> 16) | (1u << 16));           // tensor_dim0[31:16], dim1=1
  g1[3] = (int)((total & 0xFFFFu) << 16);              // tile_dim0 = total (<65536)
  g1[4] = 1;                                           // tile_dim1 = 1
  g1[5] = (int)total;                                  // tensor_dim0_stride[31:0]
  g1[6] = 0;
  g1[7] = 0;
  v4i z4 = {0, 0, 0, 0};
  v8i z8 = {0, 0, 0, 0, 0, 0, 0, 0};
  __builtin_amdgcn_tensor_load_to_lds(g0, g1, z4, z4, z8, 0);
}
#endif

// ---------------------------------------------------------------------------
// conv0: 5x5 stride-2, pad (top/left 1, bottom/right 2), 3->24 ch, ReLU.
// ---------------------------------------------------------------------------
__global__ void conv0_kernel(const float* __restrict__ x,
                             const float* __restrict__ w,
                             const float* __restrict__ b,
                             float* __restrict__ y, int N) {
  int idx = blockIdx.x * blockDim.x + threadIdx.x;
  int total = N * 24 * 64 * 64;
  if (idx >= total) return;
  int ox = idx & 63;
  int oy = (idx >> 6) & 63;
  int co = (idx >> 12) % 24;
  int n  = idx / (24 * 64 * 64);
  const float* xn = x + (size_t)n * 3 * 128 * 128;
  const float* wc = w + co * 3 * 25;
  float acc = b[co];
  for (int ci = 0; ci < 3; ++ci) {
    #pragma unroll
    for (int ky = 0; ky < 5; ++ky) {
      int iy = oy * 2 + ky - 1;
      if ((unsigned)iy >= 128u) continue;
      #pragma unroll
      for (int kx = 0; kx < 5; ++kx) {
        int ix = ox * 2 + kx - 1;
        if ((unsigned)ix >= 128u) continue;
        acc = fmaf(xn[ci * 128 * 128 + iy * 128 + ix], wc[ci * 25 + ky * 5 + kx], acc);
      }
    }
  }
  y[idx] = fmaxf(acc, 0.f);
}

// ---------------------------------------------------------------------------
// depthwise 3x3. stride 1: SAME pad (1,1). stride 2: pad (0,2) H and W.
// ---------------------------------------------------------------------------
__global__ void dw3x3_kernel(const float* __restrict__ x,
                             const float* __restrict__ w,
                             const float* __restrict__ b,
                             float* __restrict__ y,
                             int N, int C, int H, int stride) {
  int OH = (stride == 2) ? (H >> 1) : H;
  int idx = blockIdx.x * blockDim.x + threadIdx.x;
  int total = N * C * OH * OH;
  if (idx >= total) return;
  int ox = idx % OH;
  int oy = (idx / OH) % OH;
  int c  = (idx / (OH * OH)) % C;
  int n  = idx / (OH * OH * C);
  const float* xn = x + ((size_t)n * C + c) * H * H;
  const float* wc = w + c * 9;
  float acc = b[c];
  #pragma unroll
  for (int ky = 0; ky < 3; ++ky) {
    int iy = (stride == 2) ? (oy * 2 + ky) : (oy + ky - 1);
    if ((unsigned)iy >= (unsigned)H) continue;
    #pragma unroll
    for (int kx = 0; kx < 3; ++kx) {
      int ix = (stride == 2) ? (ox * 2 + kx) : (ox + kx - 1);
      if ((unsigned)ix >= (unsigned)H) continue;
      acc = fmaf(xn[iy * H + ix], wc[ky * 3 + kx], acc);
    }
  }
  y[idx] = acc;
}

// ---------------------------------------------------------------------------
// 2x2/2 max pool (skip branch of stride-2 blocks).
// ---------------------------------------------------------------------------
__global__ void maxpool2_kernel(const float* __restrict__ x, float* __restrict__ y,
                                int N, int C, int H) {
  int OH = H >> 1;
  int idx = blockIdx.x * blockDim.x + threadIdx.x;
  int total = N * C * OH * OH;
  if (idx >= total) return;
  int ox = idx % OH;
  int oy = (idx / OH) % OH;
  int c  = (idx / (OH * OH)) % C;
  int n  = idx / (OH * OH * C);
  const float* xn = x + ((size_t)n * C + c) * H * H;
  int iy = oy * 2, ix = ox * 2;
  float m = fmaxf(fmaxf(xn[iy * H + ix], xn[iy * H + ix + 1]),
                  fmaxf(xn[(iy + 1) * H + ix], xn[(iy + 1) * H + ix + 1]));
  y[idx] = m;
}

// ---------------------------------------------------------------------------
// Pointwise 1x1 conv as fp32 WMMA GEMM, fused bias + channel-padded residual +
// ReLU. Fully specialized on (CIN, COUT, HW): loops unroll, all offsets become
// immediates, no EXEC churn in the hot loop.
//   - Weights: raw [COUT][CIN] fp32 in LDS, staged by the Tensor Data Mover
//     (one wave issues tensor_load_to_lds; s_wait_tensorcnt 0; barrier).
//     Rows >= COUT of boundary M-tiles read garbage LDS -> only corrupt D rows
//     that are never stored (WMMA rows are independent).
//   - A operand: ds_load_b64 per (M-tile, K-step). B operand: two coalesced
//     global_load_b32 per K-step, loaded ONCE and reused across all M-tiles
//     (K-outer / M-inner with v8f acc[MT]).
//   - Wave = one 16-pixel tile, full EXEC at every WMMA (tail guard is
//     wave-uniform).
// ---------------------------------------------------------------------------
template <int CIN, int COUT, int HW>
__global__ void __launch_bounds__(256)
pw_wmma_kernel(const float* __restrict__ X, const float* __restrict__ Wg,
               const float* __restrict__ Bg, const float* __restrict__ S,
               float* __restrict__ Y, int total_tiles) {
  constexpr int MT = (COUT + 15) / 16;       // M tiles over cout
  constexpr int KSTEPS = (CIN + 3) / 4;      // K steps of 4
  __shared__ float lds_w[MT * 16 * CIN + 4]; // +4: clamped K-tail reads stay in-bounds
  __shared__ float lds_b[96];

#if HAVE_TDM
  if ((threadIdx.x >> 5) == 0) {
    tdm_load_1d(Wg, &lds_w[0], (unsigned)(COUT * CIN));
  }
#else
  for (int i = threadIdx.x; i < COUT * CIN; i += 256) lds_w[i] = Wg[i];
#endif
  for (int i = threadIdx.x; i < COUT; i += 256) lds_b[i] = Bg[i];
#if HAVE_TDM
  if ((threadIdx.x >> 5) == 0) __builtin_amdgcn_s_wait_tensorcnt(0);
#endif
  __syncthreads();

  int wave = threadIdx.x >> 5;               // 8 waves / block
  int lane = threadIdx.x & 31;
  int t = blockIdx.x * 8 + wave;             // one 16-pixel tile per wave
  if (t >= total_tiles) return;              // wave-uniform exit

  constexpr int TPI = HW / 16;               // HW always a multiple of 16
  int n  = t / TPI;
  int p0 = (t - n * TPI) * 16;

  int lanelo = lane & 15;                    // pixel (N) / weight-row (M) index
  int half   = lane >> 4;                    // selects K pair {0,1} vs {2,3} / M+8
  int koff   = half * 2;
  int col    = p0 + lanelo;

  const float* Xn = X + (size_t)n * CIN * HW + col;
  const float* Sn = S + (size_t)n * CIN * HW + col;
  float*       Yn = Y + (size_t)n * COUT * HW + col;
  const float* Arow = &lds_w[lanelo * CIN];  // A: M striped over lanes 0..15

  v8f acc[MT] = {};
  #pragma unroll
  for (int ks = 0; ks < KSTEPS; ++ks) {
    int ka = ks * 4 + koff;                  // A/B layout: v0 = K{0,2}, v1 = K{1,3}
    v2f bm;
    if (ks * 4 + 3 < CIN) {                  // folds per unrolled step
      bm.x = Xn[(size_t)ka * HW];
      bm.y = Xn[(size_t)(ka + 1) * HW];
    } else {                                 // K tail (only the CIN=42 layer)
      float b0 = Xn[(size_t)(ka < CIN ? ka : 0) * HW];
      float b1 = Xn[(size_t)(ka + 1 < CIN ? ka + 1 : 0) * HW];
      bm.x = (ka < CIN) ? b0 : 0.f;
      bm.y = (ka + 1 < CIN) ? b1 : 0.f;
    }
    #pragma unroll
    for (int mt = 0; mt < MT; ++mt) {
      v2f a;
      const float* wr = Arow + mt * 16 * CIN;
      if (ks * 4 + 3 < CIN) {
        a.x = wr[ka];                        // contiguous pair -> ds_load_b64
        a.y = wr[ka + 1];
      } else {
        float a0 = wr[ka];                   // in-bounds via +4 pad
        float a1 = wr[ka + 1];
        a.x = (ka < CIN) ? a0 : 0.f;
        a.y = (ka + 1 < CIN) ? a1 : 0.f;
      }
      acc[mt] = __builtin_amdgcn_wmma_f32_16x16x4_f32(false, a, false, bm,
                                                      (short)0, acc[mt],
                                                      false, false);
    }
  }

  // C/D layout: VGPR r -> M = r (lanes 0-15) / r+8 (lanes 16-31), N = lanelo.
  #pragma unroll
  for (int mt = 0; mt < MT; ++mt) {
    int mbase = mt * 16 + half * 8;
    #pragma unroll
    for (int r = 0; r < 8; ++r) {
      int m = mbase + r;
      if ((mt * 16 + 15 < COUT) || (m < COUT)) {   // folds for interior tiles
        float v = acc[mt][r] + lds_b[m];
        if ((mt * 16 + 15 < CIN) || (m < CIN))     // channel-padded residual
          v += Sn[(size_t)m * HW];
        Yn[(size_t)m * HW] = fmaxf(v, 0.f);
      }
    }
  }
}

// ---------------------------------------------------------------------------
// Head: 1x1 conv, NHWC flatten into the (r, c) output slabs.
// ---------------------------------------------------------------------------
__global__ void head_kernel(const float* __restrict__ F, const float* __restrict__ Wg,
                            const float* __restrict__ Bg, float* __restrict__ out,
                            int N, int cin, int cout, int HW,
                            int row_stride, int offset) {
  int idx = blockIdx.x * blockDim.x + threadIdx.x;
  int total = N * HW * cout;
  if (idx >= total) return;
  int co = idx % cout;
  int p  = (idx / cout) % HW;
  int n  = idx / (cout * HW);
  const float* f = F + (size_t)n * cin * HW + p;
  const float* w = Wg + co * cin;
  float acc = Bg[co];
  for (int k = 0; k < cin; ++k) acc = fmaf(f[(size_t)k * HW], w[k], acc);
  out[(size_t)n * row_stride + offset + p * cout + co] = acc;
}

// ---------------------------------------------------------------------------
// One BlazeBlock: depthwise (+maxpool skip for s2) then WMMA pointwise.
// ---------------------------------------------------------------------------
template <int CIN, int COUT, int STRIDE, int HIN>
static void run_block(const float* cur, float* dwbuf, float* skipbuf, float* nxt,
                      void* const* d_in, int pidx, int N, hipStream_t stream) {
  constexpr int OH = (STRIDE == 2) ? HIN / 2 : HIN;
  constexpr int HW = OH * OH;
  const float* dww = (const float*)d_in[pidx + 0];
  const float* dwb = (const float*)d_in[pidx + 1];
  const float* pww = (const float*)d_in[pidx + 2];
  const float* pwb = (const float*)d_in[pidx + 3];
  int total = N * CIN * HW;
  dw3x3_kernel<<<cdiv_h(total, 256), 256, 0, stream>>>(cur, dww, dwb, dwbuf,
                                                       N, CIN, HIN, STRIDE);
  const float* skip = cur;
  if (STRIDE == 2) {
    maxpool2_kernel<<<cdiv_h(total, 256), 256, 0, stream>>>(cur, skipbuf, N, CIN, HIN);
    skip = skipbuf;
  }
  int tiles = (N * HW) / 16;
  pw_wmma_kernel<CIN, COUT, HW><<<cdiv_h(tiles, 8), 256, 0, stream>>>(
      dwbuf, pww, pwb, skip, nxt, tiles);
}

// ---------------------------------------------------------------------------
// Host orchestration
// ---------------------------------------------------------------------------
extern "C" void kernel_launch(void* const* d_in, const int* in_sizes, int n_in,
                              void* d_out, int out_size, void* d_ws, size_t ws_size,
                              hipStream_t stream) {
  (void)in_sizes; (void)n_in; (void)out_size; (void)ws_size;
  const int N = 128;
  const float* x   = (const float*)d_in[0];
  const float* c0w = (const float*)d_in[1];
  const float* c0b = (const float*)d_in[2];

  // ---- workspace layout (floats), ~180 MB total ----
  float* ws = (float*)d_ws;
  size_t off = 0;
  float* bufA    = ws + off; off += 14680064;  // max act: 128*28*64*64
  float* bufB    = ws + off; off += 14680064;
  float* dwbuf   = ws + off; off += 12582912;  // max dw out: 128*24*64*64
  float* skipbuf = ws + off; off += 3670016;   // max pool out: 128*28*32*32
  float* bufC    = ws + off; off += 786432;    // bb2 ping: 128*96*8*8
  float* bufD    = ws + off; off += 786432;    // bb2 pong

  // conv0 -> bufA [N,24,64,64]
  {
    int total = N * 24 * 64 * 64;
    conv0_kernel<<<cdiv_h(total, 256), 256, 0, stream>>>(x, c0w, c0b, bufA, N);
  }

  // bb1 (params at d_in[3+4i])
  run_block<24, 24, 1, 64>(bufA, dwbuf, skipbuf, bufB, d_in,  3, N, stream);
  run_block<24, 28, 1, 64>(bufB, dwbuf, skipbuf, bufA, d_in,  7, N, stream);
  run_block<28, 32, 2, 64>(bufA, dwbuf, skipbuf, bufB, d_in, 11, N, stream);
  run_block<32, 36, 1, 32>(bufB, dwbuf, skipbuf, bufA, d_in, 15, N, stream);
  run_block<36, 42, 1, 32>(bufA, dwbuf, skipbuf, bufB, d_in, 19, N, stream);
  run_block<42, 48, 2, 32>(bufB, dwbuf, skipbuf, bufA, d_in, 23, N, stream);
  run_block<48, 56, 1, 16>(bufA, dwbuf, skipbuf, bufB, d_in, 27, N, stream);
  run_block<56, 64, 1, 16>(bufB, dwbuf, skipbuf, bufA, d_in, 31, N, stream);
  run_block<64, 72, 1, 16>(bufA, dwbuf, skipbuf, bufB, d_in, 35, N, stream);
  run_block<72, 80, 1, 16>(bufB, dwbuf, skipbuf, bufA, d_in, 39, N, stream);
  run_block<80, 88, 1, 16>(bufA, dwbuf, skipbuf, bufB, d_in, 43, N, stream);
  float* feat8 = bufB;  // [N,88,16,16]; bb2 runs in bufC/bufD and never touches it

  // bb2 (params at d_in[47+4i])
  run_block<88, 96, 2, 16>(bufB, dwbuf, skipbuf, bufC, d_in, 47, N, stream);
  run_block<96, 96, 1,  8>(bufC, dwbuf, skipbuf, bufD, d_in, 51, N, stream);
  run_block<96, 96, 1,  8>(bufD, dwbuf, skipbuf, bufC, d_in, 55, N, stream);
  run_block<96, 96, 1,  8>(bufC, dwbuf, skipbuf, bufD, d_in, 59, N, stream);
  run_block<96, 96, 1,  8>(bufD, dwbuf, skipbuf, bufC, d_in, 63, N, stream);
  float* feat16 = bufC; // [N,96,8,8]

  // heads
  const float* c8w  = (const float*)d_in[67];
  const float* c8b  = (const float*)d_in[68];
  const float* c16w = (const float*)d_in[69];
  const float* c16b = (const float*)d_in[70];
  const float* r8w  = (const float*)d_in[71];
  const float* r8b  = (const float*)d_in[72];
  const float* r16w = (const float*)d_in[73];
  const float* r16b = (const float*)d_in[74];

  float* r_out = (float*)d_out;                         // [N,896,16]
  float* c_out = (float*)d_out + (size_t)N * 896 * 16;  // [N,896,1]

  { int total = N * 256 * 32;
    head_kernel<<<cdiv_h(total, 256), 256, 0, stream>>>(feat8,  r8w,  r8b,  r_out,
                                                        N, 88, 32, 256, 896 * 16, 0); }
  { int total = N * 64 * 96;
    head_kernel<<<cdiv_h(total, 256), 256, 0, stream>>>(feat16, r16w, r16b, r_out,
                                                        N, 96, 96, 64, 896 * 16, 8192); }
  { int total = N * 256 * 2;
    head_kernel<<<cdiv_h(total, 256), 256, 0, stream>>>(feat8,  c8w,  c8b,  c_out,
                                                        N, 88, 2, 256, 896, 0); }
  { int total = N * 64 * 6;
    head_kernel<<<cdiv_h(total, 256), 256, 0, stream>>>(feat16, c16w, c16b, c_out,
                                                        N, 96, 6, 64, 896, 512); }
}